// ImprovedClusteredAttention_86071144612551
// MI455X (gfx1250) — compile-verified
//
#include <hip/hip_runtime.h>

// Problem constants (from reference)
#define N_ 2
#define L_ 2048
#define H_ 8
#define E_ 64
#define S_ 2048
#define C_ 256
#define ITER_ 10
#define BITS_ 32
#define TOPK_ 32
#define NH_ (N_*H_)
#define TSCALE 0.125f  // 1/sqrt(64)

typedef float v2f __attribute__((ext_vector_type(2)));
typedef float v8f __attribute__((ext_vector_type(8)));

// ---------------------------------------------------------------------------
// Kernel 1: LSH hash -> one packed uint32 code per query.
// proj_b = dot(Q, planes[b,:64]) + planes[b,64];  bit b = (proj_b > 0)
// ---------------------------------------------------------------------------
__global__ __launch_bounds__(256)
void k_hash(const float* __restrict__ Q, const float* __restrict__ planes,
            unsigned* __restrict__ codes) {
  int t = blockIdx.x * blockDim.x + threadIdx.x;      // (n,h,l) flattened
  if (t >= NH_ * L_) return;
  int l = t % L_;
  int nh = t / L_;
  int h = nh % H_, n = nh / H_;
  const float* q = Q + (((size_t)n * L_ + l) * H_ + h) * E_;
  float acc[BITS_];
#pragma unroll
  for (int b = 0; b < BITS_; ++b) acc[b] = planes[b * (E_ + 1) + E_];  // bias
  for (int e = 0; e < E_; ++e) {
    float qv = q[e];
#pragma unroll
    for (int b = 0; b < BITS_; ++b) acc[b] += qv * planes[b * (E_ + 1) + e];
  }
  unsigned code = 0;
#pragma unroll
  for (int b = 0; b < BITS_; ++b) code |= (acc[b] > 0.0f) ? (1u << b) : 0u;
  codes[t] = code;
}

// ---------------------------------------------------------------------------
// Kernel 2: Hamming k-means (bit-packed). One workgroup per (n,h).
// dist = hsum + csum - 2*dot == popcount(code ^ cent) for binary vectors.
// Tie-break: first (lowest) cluster index, matching jnp.argmin.
// ---------------------------------------------------------------------------
__global__ __launch_bounds__(256)
void k_kmeans(const unsigned* __restrict__ codes,
              int* __restrict__ assign_out, float* __restrict__ counts_out) {
  __shared__ unsigned codes_s[L_];
  __shared__ unsigned cent_s[C_];
  __shared__ unsigned char assign_s[L_];
  __shared__ int cnt_s[C_];
  int nh = blockIdx.x;
  int t = threadIdx.x;
  for (int j = t; j < L_; j += 256) codes_s[j] = codes[(size_t)nh * L_ + j];
  __syncthreads();
  if (t < C_) cent_s[t] = codes_s[t * (L_ / C_)];   // init_idx = c*L/C
  __syncthreads();

  for (int it = 0; it < ITER_; ++it) {
    // assign: each thread owns 8 queries
    for (int j = 0; j < L_ / 256; ++j) {
      int l = t + 256 * j;
      unsigned code = codes_s[l];
      int bestd = 1 << 30, bestc = 0;
      for (int c = 0; c < C_; ++c) {
        int d = __popc(code ^ cent_s[c]);
        if (d < bestd) { bestd = d; bestc = c; }
      }
      assign_s[l] = (unsigned char)bestc;
    }
    __syncthreads();
    // update: thread t == cluster t; per-bit sums in unrolled registers
    {
      int cnt = 0;
      int bs[BITS_];
#pragma unroll
      for (int b = 0; b < BITS_; ++b) bs[b] = 0;
      for (int l = 0; l < L_; ++l) {
        if ((int)assign_s[l] == t) {
          ++cnt;
          unsigned code = codes_s[l];
#pragma unroll
          for (int b = 0; b < BITS_; ++b) bs[b] += (code >> b) & 1u;
        }
      }
      unsigned nc = 0;
#pragma unroll
      for (int b = 0; b < BITS_; ++b)
        if (2 * bs[b] >= cnt) nc |= 1u << b;      // sums/cnt >= 0.5
      if (cnt > 0) cent_s[t] = nc;                // keep old when empty
    }
    __syncthreads();
  }

  // final assignment + cluster counts
  if (t < C_) cnt_s[t] = 0;
  __syncthreads();
  for (int j = 0; j < L_ / 256; ++j) {
    int l = t + 256 * j;
    unsigned code = codes_s[l];
    int bestd = 1 << 30, bestc = 0;
    for (int c = 0; c < C_; ++c) {
      int d = __popc(code ^ cent_s[c]);
      if (d < bestd) { bestd = d; bestc = c; }
    }
    assign_out[(size_t)nh * L_ + l] = bestc;
    atomicAdd(&cnt_s[bestc], 1);
  }
  __syncthreads();
  if (t < C_) counts_out[(size_t)nh * C_ + t] = (float)cnt_s[t];
}

// ---------------------------------------------------------------------------
// Kernel 3: Qg[c][e] = mean of Q over cluster members (LDS float atomics).
// One workgroup per (n,h). Dynamic LDS: C_*E_ floats (64 KB).
// ---------------------------------------------------------------------------
__global__ __launch_bounds__(256)
void k_qg(const float* __restrict__ Q, const int* __restrict__ assign,
          const float* __restrict__ counts, float* __restrict__ Qg) {
  extern __shared__ float qg_s[];
  int nh = blockIdx.x;
  int h = nh % H_, n = nh / H_;
  int t = threadIdx.x;
  for (int i = t; i < C_ * E_; i += 256) qg_s[i] = 0.0f;
  __syncthreads();
  for (int i = t; i < L_ * E_; i += 256) {
    int l = i >> 6, e = i & 63;
    int c = assign[(size_t)nh * L_ + l];
    atomicAdd(&qg_s[c * E_ + e], Q[(((size_t)n * L_ + l) * H_ + h) * E_ + e]);
  }
  __syncthreads();
  for (int i = t; i < C_ * E_; i += 256) {
    int c = i >> 6;
    float cnt = counts[(size_t)nh * C_ + c];
    Qg[(size_t)nh * C_ * E_ + i] = qg_s[i] / fmaxf(cnt, 1.0f);
  }
}

// ---------------------------------------------------------------------------
// Kernel 4: the heavy kernel. One workgroup per (nh, 16-cluster tile).
//   Phase A : QK(16x2048) = Qg(16x64) x K^T  via v_wmma_f32_16x16x4_f32
//   Phase B1: per row: max, exp-overwrite, softmax denom (wave-per-row)
//   Phase B2: Vg_all(16x64) = exp(16x2048) x V(2048x64) via WMMA
//             (4 column tiles x 2 K-halves spread over the 8 waves)
//   Phase B3: per row: exact top-32 (exp is monotonic in score -> identical
//             selection), subtract top contributions, normalize, write.
// Dynamic LDS: 16*64 (Qg) + 16*2048 (scores) + 2*16*64 (Vg partials)
//              + 16 (denoms) floats = ~140 KB (< 320 KB WGP LDS).
// ---------------------------------------------------------------------------
__global__ __launch_bounds__(256)
void k_scores(const float* __restrict__ Kk, const float* __restrict__ Vv,
              const float* __restrict__ Qg, int* __restrict__ topi,
              float* __restrict__ abk, float* __restrict__ vg) {
  extern __shared__ float lds[];
  float* qg_s   = lds;                         // 16 x 64
  float* sc     = lds + 16 * E_;               // 16 x 2048
  float* vgp    = sc + 16 * S_;                // 2 x 16 x 64 (K-half partials)
  float* denom_s= vgp + 2 * 16 * E_;           // 16
  int blk = blockIdx.x;
  int nh = blk >> 4, mt = blk & 15;
  int h = nh % H_, n = nh / H_;
  int cbase = mt * 16;
  int t = threadIdx.x;
  int lane = t & 31, wave = t >> 5;
  int half = lane >> 4, m = lane & 15;

  for (int i = t; i < 16 * E_; i += 256)
    qg_s[i] = Qg[(size_t)nh * C_ * E_ + (size_t)cbase * E_ + i];
  __syncthreads();

  // ---- Phase A: QK via WMMA. A lane layout (16x4 f32): lane m = row,
  // VGPRj = K = 2*half + j.  B (4x16): lane m = col, VGPRj = K = 2*half + j.
  for (int st = wave; st < S_ / 16; st += 8) {
    int sb = st * 16;
    v8f acc = {};
    const float* krow = Kk + (((size_t)n * S_ + sb + m) * H_ + h) * E_;
#pragma unroll
    for (int kb = 0; kb < E_; kb += 4) {
      v2f a = *(const v2f*)&qg_s[m * E_ + kb + 2 * half];
      v2f b = *(const v2f*)&krow[kb + 2 * half];
      acc = __builtin_amdgcn_wmma_f32_16x16x4_f32(
          false, a, false, b, (short)0, acc, false, false);
    }
    // D layout: VGPR v -> row (v + 8*half), col (lane&15)
#pragma unroll
    for (int v = 0; v < 8; ++v)
      sc[(v + 8 * half) * S_ + sb + m] = acc[v];
  }
  __syncthreads();

  // ---- Phase B1: row max + exp overwrite + softmax denominator
  for (int r = wave; r < 16; r += 8) {
    float* row = sc + (size_t)r * S_;
    float mx = -3.4e38f;
    for (int i = lane; i < S_; i += 32) mx = fmaxf(mx, row[i]);
    for (int off = 16; off > 0; off >>= 1) mx = fmaxf(mx, __shfl_xor(mx, off, 32));
    float sum = 0.0f;
    for (int i = lane; i < S_; i += 32) {
      float e = __expf(TSCALE * (row[i] - mx));
      row[i] = e;
      sum += e;
    }
    for (int off = 16; off > 0; off >>= 1) sum += __shfl_xor(sum, off, 32);
    if (lane == 0) denom_s[r] = sum;
  }
  __syncthreads();

  // ---- Phase B2: Vg_all = exp(16x2048) x V(2048x64) via WMMA.
  // wave -> (column tile ct = wave&3, K half kh = wave>>2); 256 WMMA each.
  {
    int ct = wave & 3;                    // e-tile: [ct*16, ct*16+16)
    int kh = wave >> 2;                   // K half: [kh*1024, kh*1024+1024)
    int ecol = ct * 16 + m;               // this lane's output column (e dim)
    const float* vcol = Vv + (size_t)n * S_ * H_ * E_ + (size_t)h * E_ + ecol;
    v8f acc = {};
    for (int kb = kh * (S_ / 2); kb < kh * (S_ / 2) + S_ / 2; kb += 4) {
      int s0 = kb + 2 * half;
      v2f a = *(const v2f*)&sc[m * S_ + s0];        // A: row m, K = s0, s0+1
      v2f b;
      b.x = vcol[(size_t)s0 * (H_ * E_)];           // B: V[s0][ecol]
      b.y = vcol[(size_t)(s0 + 1) * (H_ * E_)];     //    V[s0+1][ecol]
      acc = __builtin_amdgcn_wmma_f32_16x16x4_f32(
          false, a, false, b, (short)0, acc, false, false);
    }
#pragma unroll
    for (int v = 0; v < 8; ++v)
      vgp[kh * 16 * E_ + (v + 8 * half) * E_ + ecol] = acc[v];
  }
  __syncthreads();

  // ---- Phase B3: exact top-32 per row, subtract, normalize, write
  for (int r = wave; r < 16; r += 8) {
    float* row = sc + (size_t)r * S_;
    float sum = denom_s[r];
    int e0 = lane, e1 = lane + 32;
    float va = vgp[r * E_ + e0] + vgp[16 * E_ + r * E_ + e0];
    float vb = vgp[r * E_ + e1] + vgp[16 * E_ + r * E_ + e1];
    float topsum = 0.0f;
    size_t tibase = ((size_t)nh * C_ + cbase + r) * TOPK_;
    for (int it = 0; it < TOPK_; ++it) {
      float bv = -1.0f;
      int bi = 0x7fffffff;
      for (int i = lane; i < S_; i += 32) {
        float v = row[i];
        if (v > bv) { bv = v; bi = i; }
      }
      for (int off = 1; off < 32; off <<= 1) {
        float ov = __shfl_xor(bv, off, 32);
        int oi = __shfl_xor(bi, off, 32);
        if (ov > bv || (ov == bv && oi < bi)) { bv = ov; bi = oi; }
      }
      if (lane == 0) { topi[tibase + it] = bi; row[bi] = -1.0f; }
      topsum += bv;
      const float* vr = Vv + (((size_t)n * S_ + bi) * H_ + h) * E_;
      va -= bv * vr[e0];
      vb -= bv * vr[e1];
    }
    if (lane == 0) abk[(size_t)nh * C_ + cbase + r] = (sum - topsum) / sum;
    float* vgrow = vg + ((size_t)nh * C_ + cbase + r) * E_;
    vgrow[e0] = va / sum;
    vgrow[e1] = vb / sum;
  }
}

// ---------------------------------------------------------------------------
// Kernel 5: per-query top-32 attention + bottom (cluster) residual.
// One wave per query; lane k handles top key k, then lanes switch to
// output dims (lane, lane+32) for the V_top accumulation.
// ---------------------------------------------------------------------------
__global__ __launch_bounds__(256)
void k_out(const float* __restrict__ Q, const float* __restrict__ Kk,
           const float* __restrict__ Vv, const int* __restrict__ assign,
           const int* __restrict__ topi, const float* __restrict__ abk,
           const float* __restrict__ vg, float* __restrict__ out) {
  __shared__ float at_s[8][TOPK_];
  __shared__ int idx_s[8][TOPK_];
  int wave = threadIdx.x >> 5, lane = threadIdx.x & 31;
  int q = blockIdx.x * 8 + wave;      // (n,h,l) flattened
  int l = q % L_;
  int nh = q / L_;
  int h = nh % H_, n = nh / H_;
  int c = assign[(size_t)nh * L_ + l];
  const float* qrow = Q + (((size_t)n * L_ + l) * H_ + h) * E_;
  int ki = topi[((size_t)nh * C_ + c) * TOPK_ + lane];
  const float* krow = Kk + (((size_t)n * S_ + ki) * H_ + h) * E_;
  __builtin_prefetch(krow, 0, 1);
  float dot = 0.0f;
  for (int e = 0; e < E_; ++e) dot += qrow[e] * krow[e];
  float mx = dot;
  for (int off = 16; off > 0; off >>= 1) mx = fmaxf(mx, __shfl_xor(mx, off, 32));
  float ex = __expf(TSCALE * (dot - mx));
  float s = ex;
  for (int off = 16; off > 0; off >>= 1) s += __shfl_xor(s, off, 32);
  float scale = 1.0f - abk[(size_t)nh * C_ + c];
  at_s[wave][lane] = ex * (scale / s);
  idx_s[wave][lane] = ki;
  asm volatile("" ::: "memory");   // order LDS store before reads (same wave)
  int e0 = lane, e1 = lane + 32;
  const float* vgrow = vg + ((size_t)nh * C_ + c) * E_;
  float a0 = vgrow[e0], a1 = vgrow[e1];
  for (int k = 0; k < TOPK_; ++k) {
    float w = at_s[wave][k];
    const float* vr = Vv + (((size_t)n * S_ + idx_s[wave][k]) * H_ + h) * E_;
    a0 += w * vr[e0];
    a1 += w * vr[e1];
  }
  float* orow = out + (((size_t)n * L_ + l) * H_ + h) * E_;   // (N,L,H,E)
  orow[e0] = a0;
  orow[e1] = a1;
}

// ---------------------------------------------------------------------------
extern "C" void kernel_launch(void* const* d_in, const int* in_sizes, int n_in,
                              void* d_out, int out_size, void* d_ws, size_t ws_size,
                              hipStream_t stream) {
  (void)in_sizes; (void)n_in; (void)out_size; (void)ws_size;
  const float* Q      = (const float*)d_in[0];  // (N,L,H,E)
  const float* Kk     = (const float*)d_in[1];  // (N,S,H,E)
  const float* Vv     = (const float*)d_in[2];  // (N,S,H,E)
  const float* planes = (const float*)d_in[3];  // (BITS,E+1)
  float* out = (float*)d_out;

  // workspace layout (bytes)
  char* ws = (char*)d_ws;
  unsigned* codes = (unsigned*)(ws + 0);            // NH*L u32      = 128 KB
  int* assign     = (int*)(ws + 131072);            // NH*L i32      = 128 KB
  float* counts   = (float*)(ws + 262144);          // NH*C f32      =  16 KB
  float* Qg       = (float*)(ws + 278528);          // NH*C*E f32    =   1 MB
  int* topi       = (int*)(ws + 1327104);           // NH*C*32 i32   = 512 KB
  float* abk      = (float*)(ws + 1851392);         // NH*C f32      =  16 KB
  float* vg       = (float*)(ws + 1867776);         // NH*C*E f32    =   1 MB

  size_t lds4 = (size_t)(16 * E_ + 16 * S_ + 2 * 16 * E_ + 16) * sizeof(float);
  k_hash  <<<(NH_ * L_ + 255) / 256, 256, 0, stream>>>(Q, planes, codes);
  k_kmeans<<<NH_, 256, 0, stream>>>(codes, assign, counts);
  k_qg    <<<NH_, 256, (size_t)C_ * E_ * sizeof(float), stream>>>(Q, assign, counts, Qg);
  k_scores<<<NH_ * 16, 256, lds4, stream>>>(Kk, Vv, Qg, topi, abk, vg);
  k_out   <<<NH_ * L_ / 8, 256, 0, stream>>>(Q, Kk, Vv, assign, topi, abk, vg, out);
}